// FusionModule_70806830842484
// MI455X (gfx1250) — compile-verified
//
#include <hip/hip_runtime.h>

// FusionModule: overlap-add gather + count-normalize + LayerNorm(D=256).
// HBM-bound (~200MB -> ~8.6us @ 23.3TB/s). Gather formulation: each output
// position l is fed by <=3 nodes; each input element is read exactly once.
// LayerNorm row reductions are done on the matrix core with the exact-f32
// V_WMMA_F32_16X16X4_F32 (A = fused tile chunks, B = ones).

typedef __attribute__((ext_vector_type(2))) float v2f;
typedef __attribute__((ext_vector_type(8))) float v8f;

#define BATCH   8
#define NNODES  64
#define PLEN    256
#define EMBED   256
#define TPB     256
#define TILE_ROWS 16
#define LDS_STRIDE 260   // 256 + 4 pad: WMMA A-fetch hits all 64 banks once

__global__ __launch_bounds__(TPB)
void fusion_overlapadd_ln_kernel(const float* __restrict__ x,      // [B,N,P,D]
                                 const float* __restrict__ gamma,  // [D]
                                 const float* __restrict__ beta,   // [D]
                                 float* __restrict__ out,          // [B,L,D]
                                 int L, int step, int nRows)
{
    __shared__ float fused[TILE_ROWS * LDS_STRIDE];
    __shared__ float pSum[8][TILE_ROWS];
    __shared__ float pSq [8][TILE_ROWS];
    __shared__ float sMu[TILE_ROWS];
    __shared__ float sRs[TILE_ROWS];

    const int tid    = threadIdx.x;
    const int wave   = tid >> 5;     // 0..7
    const int lane   = tid & 31;
    const int rLocal = tid >> 4;     // 0..15: row handled in phase 1/3
    const int q      = tid & 15;     // col group within the row

    const int rowBase = blockIdx.x * TILE_ROWS;

    // ---------------- Phase 1: gather + overlap-add + count-normalize -> LDS
    {
        int row = rowBase + rLocal;
        if (row >= nRows) row = nRows - 1;          // clamp; keeps EXEC full
        const int b = row / L;
        const int l = row - b * L;

        int n_hi = l / step;
        if (n_hi > NNODES - 1) n_hi = NNODES - 1;
        const int p0 = l - n_hi * step;             // p0 >= 0 by construction

        const float* base = x + (size_t)b * NNODES * PLEN * EMBED;
        const float* s0 = base + ((size_t)n_hi * PLEN + p0) * (size_t)EMBED;
        // contributor n_hi-1 sits at p0+step; n_hi-2 at p0+2*step
        const long  d1 = ((long)step - (long)PLEN) * (long)EMBED;
        const float* s1 = s0 + d1;
        const float* s2 = s0 + 2 * d1;

        const bool ok0 = (p0 < PLEN);
        const bool ok1 = (n_hi - 1 >= 0) && (p0 + step     < PLEN);
        const bool ok2 = (n_hi - 2 >= 0) && (p0 + 2 * step < PLEN);
        const int  cnt = (int)ok0 + (int)ok1 + (int)ok2;
        const float inv = 1.0f / fmaxf((float)cnt, 1e-8f);

        #pragma unroll
        for (int kk = 0; kk < 4; ++kk) {
            const int col = kk * 64 + q * 4;
            float4 acc = make_float4(0.f, 0.f, 0.f, 0.f);
            if (ok0) { float4 v = *(const float4*)(s0 + col);
                       acc.x += v.x; acc.y += v.y; acc.z += v.z; acc.w += v.w; }
            if (ok1) { float4 v = *(const float4*)(s1 + col);
                       acc.x += v.x; acc.y += v.y; acc.z += v.z; acc.w += v.w; }
            if (ok2) { float4 v = *(const float4*)(s2 + col);
                       acc.x += v.x; acc.y += v.y; acc.z += v.z; acc.w += v.w; }
            acc.x *= inv; acc.y *= inv; acc.z *= inv; acc.w *= inv;
            *(float4*)&fused[rLocal * LDS_STRIDE + col] = acc;
        }
    }
    __syncthreads();

    // ---------------- Phase 2: WMMA f32 row-sum / row-sumsq reduction
    // Each wave reduces cols [32*wave, 32*wave+32): 8 chunks of K=4.
    {
        const int m  = lane & 15;          // A row held by this lane
        const int kb = (lane >> 4) * 2;    // K base: 0 (lanes 0-15) / 2 (16-31)
        const float* aptr = &fused[m * LDS_STRIDE + wave * 32 + kb];

        v8f cS = {};
        v8f cQ = {};
        const v2f ones = {1.0f, 1.0f};

        #pragma unroll
        for (int c = 0; c < 8; ++c) {
            v2f a = *(const v2f*)(aptr + 4 * c);   // ds_load_b64, conflict-free
            v2f a2; a2.x = a.x * a.x; a2.y = a.y * a.y;
            // D[m][n] += sum_k A[m][k] * 1  -> row-chunk sums broadcast over n
            cS = __builtin_amdgcn_wmma_f32_16x16x4_f32(
                    false, a,  false, ones, (short)0, cS, false, false);
            cQ = __builtin_amdgcn_wmma_f32_16x16x4_f32(
                    false, a2, false, ones, (short)0, cQ, false, false);
        }

        // Column 0 of C: lane 0 has rows 0..7 (VGPR j = row j),
        // lane 16 has rows 8..15.
        if ((lane & 15) == 0) {
            const int r0 = (lane >> 4) * 8;
            #pragma unroll
            for (int j = 0; j < 8; ++j) {
                pSum[wave][r0 + j] = cS[j];
                pSq [wave][r0 + j] = cQ[j];
            }
        }
    }
    __syncthreads();

    if (tid < TILE_ROWS) {
        float s = 0.f, ss = 0.f;
        #pragma unroll
        for (int w = 0; w < 8; ++w) { s += pSum[w][tid]; ss += pSq[w][tid]; }
        const float mu  = s * (1.0f / (float)EMBED);
        float var = ss * (1.0f / (float)EMBED) - mu * mu;
        var = fmaxf(var, 0.0f);
        sMu[tid] = mu;
        sRs[tid] = rsqrtf(var + 1e-5f);
    }
    __syncthreads();

    // ---------------- Phase 3: normalize + affine + coalesced store
    {
        const int row = rowBase + rLocal;
        if (row < nRows) {
            const float mu = sMu[rLocal];
            const float rs = sRs[rLocal];
            float* orow = out + (size_t)row * EMBED;
            #pragma unroll
            for (int kk = 0; kk < 4; ++kk) {
                const int col = kk * 64 + q * 4;
                float4 v = *(const float4*)&fused[rLocal * LDS_STRIDE + col];
                float4 g = *(const float4*)(gamma + col);
                float4 t = *(const float4*)(beta + col);
                float4 o;
                o.x = (v.x - mu) * rs * g.x + t.x;
                o.y = (v.y - mu) * rs * g.y + t.y;
                o.z = (v.z - mu) * rs * g.z + t.z;
                o.w = (v.w - mu) * rs * g.w + t.w;
                *(float4*)(orow + col) = o;
            }
        }
    }
}

extern "C" void kernel_launch(void* const* d_in, const int* in_sizes, int n_in,
                              void* d_out, int out_size, void* d_ws, size_t ws_size,
                              hipStream_t stream)
{
    const float* x     = (const float*)d_in[0];
    const float* gamma = (const float*)d_in[1];
    const float* beta  = (const float*)d_in[2];
    // d_in[3] is seq_len (device int scalar); derive L host-side from out_size
    // so grid sizing stays graph-capture safe.
    const int L = out_size / (BATCH * EMBED);

    int step;
    if (NNODES == 1)      step = 1;
    else if (L <= PLEN)   step = 1;
    else {
        step = (L - PLEN) / (NNODES - 1);
        if (step < 1) step = 1;
    }

    const int nRows = BATCH * L;
    const int grid  = (nRows + TILE_ROWS - 1) / TILE_ROWS;

    fusion_overlapadd_ln_kernel<<<grid, TPB, 0, stream>>>(
        x, gamma, beta, (float*)d_out, L, step, nRows);
}